// TensorGraphConvolution_55490977464947
// MI455X (gfx1250) — compile-verified
//
#include <hip/hip_runtime.h>
#include <hip/hip_bf16.h>

// Model dims
#define T_DIM 16
#define N_DIM 50000
#define E_DIM 131072
#define F_DIM 64
#define BW_DIM 3

typedef __attribute__((ext_vector_type(2))) float v2f;
typedef __attribute__((ext_vector_type(8))) float v8f;

// -----------------------------------------------------------------------------
// Kernel 1: Y[k, n, :] = sum_t M[k,t] * ( x[t, n, :] @ W )
// Block = 256 threads (8 waves), each block owns a 16-wide n-tile.
// Stage 1: each wave computes Z[t] = x[t, tile, :] @ W with fp32 WMMA -> LDS.
// Stage 2: each wave computes Y[k, tile, :] = sum_t M[k,t] * Z[t] -> global.
// 50000 % 16 == 0, so every tile is full (EXEC all-1s for WMMA).
// -----------------------------------------------------------------------------
__global__ __launch_bounds__(256) void tgc_fused_transform(
    const float* __restrict__ x,   // [T, N, 64]
    const float* __restrict__ M,   // [T, T]
    const float* __restrict__ W,   // [64, 64]
    float* __restrict__ Y)         // [T, N, 64] workspace
{
    extern __shared__ float smem[];
    float* ldsW = smem;            // 64*64   = 4096 floats
    float* ldsM = smem + 4096;     // 16*16   = 256 floats
    float* ldsZ = smem + 4352;     // 16*16*64= 16384 floats (Z[t][n16][f64])

    const int tid  = threadIdx.x;
    const int lane = tid & 31;
    const int wave = tid >> 5;
    const int n0   = blockIdx.x * 16;

    // Stage weight + M into LDS (coalesced).
#pragma unroll
    for (int i = 0; i < 16; ++i) ldsW[tid + 256 * i] = W[tid + 256 * i];
    ldsM[tid] = M[tid];            // 256 == T*T
    __syncthreads();

    // ---------------- Stage 1: Z[t] = x_tile[t] @ W via V_WMMA_F32_16X16X4 ---
    // A (16x4 f32) layout: lane l -> row M = l&15 ; VGPR j -> K = 2*(l>>4)+j
    // B (4x16  f32) layout: lane l -> col N = l&15 ; VGPR j -> K = 2*(l>>4)+j
    // D (16x16 f32) layout: VGPR v -> row M = v + 8*(l>>4) ; col N = l&15
    const int row  = n0 + (lane & 15);
    const int koff = (lane >> 4) * 2;
    const int colb = lane & 15;

#pragma unroll
    for (int u = 0; u < 2; ++u) {
        const int t = wave + u * 8;
        const float* px = x + ((size_t)t * N_DIM + row) * F_DIM + koff;
#pragma unroll
        for (int fo = 0; fo < 4; ++fo) {
            v8f acc = {};
#pragma unroll
            for (int k0 = 0; k0 < 64; k0 += 4) {
                v2f a = *(const v2f*)(px + k0);
                v2f b;
                b.x = ldsW[(k0 + koff)     * 64 + fo * 16 + colb];
                b.y = ldsW[(k0 + koff + 1) * 64 + fo * 16 + colb];
                acc = __builtin_amdgcn_wmma_f32_16x16x4_f32(
                    false, a, false, b, (short)0, acc, false, false);
            }
#pragma unroll
            for (int v = 0; v < 8; ++v)
                ldsZ[t * 1024 + (v + 8 * (lane >> 4)) * 64 + fo * 16 + colb] = acc[v];
        }
    }
    __syncthreads();

    // ---------------- Stage 2: Y[k] = sum_t M[k,t] * Z[t] --------------------
    const float4* ldsZ4 = (const float4*)ldsZ;
    float4* Y4 = (float4*)Y;
#pragma unroll
    for (int u = 0; u < 2; ++u) {
        const int k = wave + u * 8;
        float4 a4[8];
#pragma unroll
        for (int i = 0; i < 8; ++i) a4[i] = make_float4(0.f, 0.f, 0.f, 0.f);
        for (int t = 0; t < 16; ++t) {
            const float m = ldsM[k * 16 + t];
#pragma unroll
            for (int i = 0; i < 8; ++i) {
                float4 z = ldsZ4[t * 256 + lane + 32 * i];
                a4[i].x += m * z.x; a4[i].y += m * z.y;
                a4[i].z += m * z.z; a4[i].w += m * z.w;
            }
        }
#pragma unroll
        for (int i = 0; i < 8; ++i) {
            const int idx  = lane + 32 * i;     // float4 index within 16x64 tile
            const int nloc = idx >> 4;
            const int f4   = idx & 15;
            Y4[((size_t)k * N_DIM + n0 + nloc) * 16 + f4] = a4[i];
        }
    }
}

// -----------------------------------------------------------------------------
// Kernel 2: banded COO scatter, fused with the M[k,k-d] band weights:
//   out[k, rows_j[e], f] += M[k,j] * vals_j[e] * Y[k, cols_j[e], f],  j = k-d
// 64 threads per edge -> contiguous 256B gather + contiguous 256B atomic adds.
// -----------------------------------------------------------------------------
__global__ __launch_bounds__(256) void tgc_scatter(
    const float* __restrict__ M,
    const float* __restrict__ vals,
    const int*   __restrict__ rows,
    const int*   __restrict__ cols,
    const float* __restrict__ Y,
    float*       __restrict__ out)
{
    const int k = blockIdx.y / BW_DIM;
    const int d = blockIdx.y % BW_DIM;
    if (k < d) return;                 // band clipped at k-d >= 0
    const int j = k - d;
    const float mkj = M[k * T_DIM + j];

    const int tid = threadIdx.x;
    const int e   = blockIdx.x * 4 + (tid >> 6);   // 4 edges per block
    const int f   = tid & 63;

    const size_t eidx = (size_t)j * E_DIM + e;
    const int   c = cols[eidx];
    const int   r = rows[eidx];
    const float coeff = mkj * vals[eidx];

    const float val = coeff * Y[((size_t)k * N_DIM + c) * F_DIM + f];
    atomicAdd(out + ((size_t)k * N_DIM + r) * F_DIM + f, val);
}

// -----------------------------------------------------------------------------
extern "C" void kernel_launch(void* const* d_in, const int* in_sizes, int n_in,
                              void* d_out, int out_size, void* d_ws, size_t ws_size,
                              hipStream_t stream) {
    const float* x        = (const float*)d_in[0];  // [T,N,64]
    const float* M        = (const float*)d_in[1];  // [T,T]
    const float* adj_vals = (const float*)d_in[2];  // [T,E]
    const float* weight   = (const float*)d_in[3];  // [64,64]
    const int*   adj_rows = (const int*)d_in[4];    // [T,E]
    const int*   adj_cols = (const int*)d_in[5];    // [T,E]

    float* Y   = (float*)d_ws;                      // [T,N,64] = 204.8 MB
    float* out = (float*)d_out;                     // [T,N,64]

    // Kernel 1: fused (M @ x) @ W  ->  Y
    const size_t smem_bytes = (size_t)(4096 + 256 + 16384) * sizeof(float); // 81 KB
    tgc_fused_transform<<<N_DIM / 16, 256, smem_bytes, stream>>>(x, M, weight, Y);

    // Zero the output accumulator (graph-capturable memset node).
    hipMemsetAsync(out, 0, (size_t)out_size * sizeof(float), stream);

    // Kernel 2: banded sparse scatter-accumulate into out.
    dim3 grid(E_DIM / 4, T_DIM * BW_DIM);
    tgc_scatter<<<grid, 256, 0, stream>>>(M, adj_vals, adj_rows, adj_cols, Y, out);
}